// HypergraphAttention_69672959476359
// MI455X (gfx1250) — compile-verified
//
#include <hip/hip_runtime.h>
#include <hip/hip_bf16.h>

typedef float v2f __attribute__((ext_vector_type(2)));
typedef float v4f __attribute__((ext_vector_type(4)));
typedef float v8f __attribute__((ext_vector_type(8)));

static constexpr int H   = 8;
static constexpr int NBV = 8 * 4096;   // B*V sites
static constexpr int D   = 256;
static constexpr float SLOPE = 0.2f;

// ---------------------------------------------------------------------------
// Kernel 1: u[k][d] = sum_e a[k*D + e] * W[e*D + d]   for k = 0,1
// Padded (16 x 256) = (16 x 256)x(256 x 256) GEMM via V_WMMA_F32_16X16X4_F32.
// Rows 2..15 of A are zeroed by a branch-free 0/1 mask (no EXEC divergence in
// the WMMA loop; WMMA requires EXEC all-ones anyway).
// 16 waves; wave w owns output columns [16w, 16w+16).
// A layout (ISA 7.12.2, 32-bit A 16x4): lane L<16 -> M=L, K={k0,k0+1};
// lane L>=16 -> M=L-16, K={k0+2,k0+3}. B 4x16 mirrors it. D 16x16: VGPR r =
// row M=r on lanes 0-15.
// ---------------------------------------------------------------------------
__global__ __launch_bounds__(512)
void hga_precompute_u(const float* __restrict__ W,
                      const float* __restrict__ a,
                      float* __restrict__ u)
{
    const int lane  = threadIdx.x & 31;
    const int wv    = threadIdx.x >> 5;    // N-tile 0..15
    const int m     = lane & 15;
    const int khalf = lane >> 4;           // 0 -> K {0,1}; 1 -> K {2,3}
    const int n     = (wv << 4) + m;

    const float amask = (m < 2) ? 1.0f : 0.0f;        // zero-pad rows 2..15
    const float* ap   = a + ((m < 2) ? m : 0) * D;    // always in-bounds

    v8f c = {};
    #pragma unroll 4
    for (int k0 = 0; k0 < D; k0 += 4) {
        const int kb = k0 + (khalf << 1);
        v2f A;
        A.x = ap[kb]     * amask;
        A.y = ap[kb + 1] * amask;
        v2f Bv;
        Bv.x = W[(size_t)kb * D + n];
        Bv.y = W[(size_t)(kb + 1) * D + n];
        c = __builtin_amdgcn_wmma_f32_16x16x4_f32(
                /*neg_a=*/false, A, /*neg_b=*/false, Bv,
                /*c_mod=*/(short)0, c, /*reuse_a=*/false, /*reuse_b=*/false);
    }
    if (lane < 16) {
        u[n]     = c[0];   // row M=0 -> u1[d]
        u[D + n] = c[1];   // row M=1 -> u2[d]
    }
}

// ---------------------------------------------------------------------------
// Kernel 2: streaming pass. One wave32 per (b,v) site; lane owns 8 d-values
// (two b128 chunks at d0=4*lane, d1=128+4*lane). All 16 NT loads for the
// site are issued as one batch (load loop separated from compute) so the
// wave keeps maximum LOADcnt in flight; Xs (256 MB) is read exactly once
// with NT hints (no reuse, exceeds 192 MB L2).
// ---------------------------------------------------------------------------
__global__ __launch_bounds__(256)
void hga_main(const float* __restrict__ Xs,
              const float* __restrict__ u,
              float* __restrict__ Y)
{
    const int lane = threadIdx.x & 31;
    const int bv   = blockIdx.x * 8 + (threadIdx.x >> 5);
    const size_t siteOff = (size_t)bv * D;
    const size_t hStride = (size_t)NBV * D;
    const int d0 = lane * 4;
    const int d1 = 128 + lane * 4;

    // per-lane chunks of u1/u2 (tiny, L2/WGP$ resident)
    const v4f u1a = *(const v4f*)(u + d0);
    const v4f u1b = *(const v4f*)(u + d1);
    const v4f u2a = *(const v4f*)(u + D + d0);
    const v4f u2b = *(const v4f*)(u + D + d1);

    // ---- issue all 16 non-temporal b128 loads up front (one clause) ----
    v4f xa[H], xb[H];
    #pragma unroll
    for (int h = 0; h < H; ++h) {
        const float* p = Xs + (size_t)h * hStride + siteOff;
        xa[h] = __builtin_nontemporal_load((const v4f*)(p + d0));
        xb[h] = __builtin_nontemporal_load((const v4f*)(p + d1));
    }

    // ---- per-h dot products + wave32 butterfly reductions ----
    float s1[H];
    float p2 = 0.0f;
    #pragma unroll
    for (int h = 0; h < H; ++h) {
        float dv1 = xa[h].x*u1a.x + xa[h].y*u1a.y + xa[h].z*u1a.z + xa[h].w*u1a.w
                  + xb[h].x*u1b.x + xb[h].y*u1b.y + xb[h].z*u1b.z + xb[h].w*u1b.w;
        float dv2 = xa[h].x*u2a.x + xa[h].y*u2a.y + xa[h].z*u2a.z + xa[h].w*u2a.w
                  + xb[h].x*u2b.x + xb[h].y*u2b.y + xb[h].z*u2b.z + xb[h].w*u2b.w;
        #pragma unroll
        for (int off = 16; off > 0; off >>= 1)
            dv1 += __shfl_xor(dv1, off, 32);
        s1[h] = dv1;
        p2 += dv2;                       // single reduction after the loop
    }
    #pragma unroll
    for (int off = 16; off > 0; off >>= 1)
        p2 += __shfl_xor(p2, off, 32);
    const float s2 = p2 * (1.0f / (float)H);   // mean over h of Xs.u2

    // ---- z -> leaky_relu -> softmax over H (registers, per lane) ----
    float z[H], zmax = -3.4e38f;
    #pragma unroll
    for (int h = 0; h < H; ++h) {
        float t = s1[h] + s2;
        z[h] = (t > 0.0f) ? t : SLOPE * t;
        zmax = fmaxf(zmax, z[h]);
    }
    float e[H], esum = 0.0f;
    #pragma unroll
    for (int h = 0; h < H; ++h) {
        e[h] = __expf(z[h] - zmax);
        esum += e[h];
    }
    const float inv = 1.0f / esum;

    // ---- weighted sum and NT store ----
    v4f ya = {0.0f, 0.0f, 0.0f, 0.0f};
    v4f yb = {0.0f, 0.0f, 0.0f, 0.0f};
    #pragma unroll
    for (int h = 0; h < H; ++h) {
        const float al = e[h] * inv;
        ya.x += al * xa[h].x; ya.y += al * xa[h].y;
        ya.z += al * xa[h].z; ya.w += al * xa[h].w;
        yb.x += al * xb[h].x; yb.y += al * xb[h].y;
        yb.z += al * xb[h].z; yb.w += al * xb[h].w;
    }
    __builtin_nontemporal_store(ya, (v4f*)(Y + siteOff + d0));
    __builtin_nontemporal_store(yb, (v4f*)(Y + siteOff + d1));
}

// ---------------------------------------------------------------------------
extern "C" void kernel_launch(void* const* d_in, const int* in_sizes, int n_in,
                              void* d_out, int out_size, void* d_ws, size_t ws_size,
                              hipStream_t stream)
{
    const float* Xs = (const float*)d_in[0];   // (8, 8, 4096, 256) f32
    const float* W  = (const float*)d_in[1];   // (256, 256) f32
    const float* a  = (const float*)d_in[2];   // (1, 512) f32
    float* Y = (float*)d_out;                  // (8, 4096, 256) f32
    float* u = (float*)d_ws;                   // 512 floats: u1 | u2

    hga_precompute_u<<<1, 512, 0, stream>>>(W, a, u);
    hga_main<<<NBV / 8, 256, 0, stream>>>(Xs, u, Y);
}